// CriticNetwork_86251533238630
// MI455X (gfx1250) — compile-verified
//
#include <hip/hip_runtime.h>

typedef float v2f __attribute__((ext_vector_type(2)));
typedef float v8f __attribute__((ext_vector_type(8)));

#define Bsz 128
#define Nn  64
#define OBS 128
#define ACT 16
#define Dd  144      // OBS+ACT
#define DK  128
#define ENC 128

#define OA_LD 148    // padded LDS row stride (floats), 592B = 16B aligned
#define W_LD  65     // padded w row stride

// workspace layout (floats): G[144*144] | u[128] | v[144]
#define G_OFF 0
#define U_OFF (Dd*Dd)
#define V_OFF (U_OFF + OBS)

// ---------------------------------------------------------------------------
// Stage 0: batch-independent precompute.
//   G[d,d'] = sum_c Wk[c,d]*Wq[c,d']      (scores = OA G OA^T)
//   u[o]    = sum_e Wenc[e,o]*Wfin[e]     (enc contribution per row)
//   v[d]    = sum_c Wv[c,d]*Wfin[128+c]   (value-vector contraction)
// ---------------------------------------------------------------------------
__global__ void precompute_kernel(const float* __restrict__ Wk,
                                  const float* __restrict__ Wq,
                                  const float* __restrict__ Wv,
                                  const float* __restrict__ Wenc,
                                  const float* __restrict__ Wfin,
                                  float* __restrict__ ws) {
  const int blk = blockIdx.x;
  const int t   = threadIdx.x;          // 0..143
  if (blk < Dd) {                       // one row of G per block
    float acc = 0.f;
    for (int c = 0; c < DK; ++c)
      acc += Wk[c * Dd + blk] * Wq[c * Dd + t];
    ws[G_OFF + blk * Dd + t] = acc;
  } else if (blk == Dd) {               // u
    if (t < OBS) {
      float acc = 0.f;
      for (int e = 0; e < ENC; ++e)
        acc += Wenc[e * OBS + t] * Wfin[e];
      ws[U_OFF + t] = acc;
    }
  } else {                              // v
    float acc = 0.f;
    for (int c = 0; c < DK; ++c)
      acc += Wv[c * Dd + t] * Wfin[ENC + c];
    ws[V_OFF + t] = acc;
  }
}

// ---------------------------------------------------------------------------
// Stage 1: one workgroup (256 threads = 8 waves) per batch element.
//   GEMM1 (WMMA f32 16x16x4): T = OA * G          [64 x 144]
//   GEMM2 (WMMA f32 16x16x4): scores = T * OA^T   [64 x 64]
//   w = sigmoid(scores/sqrt(128)); then scalar folds + streamed outputs.
// ---------------------------------------------------------------------------
__global__ void critic_kernel(const float* __restrict__ obs,
                              const float* __restrict__ pol,
                              const float* __restrict__ act,
                              const float* __restrict__ ws,
                              float* __restrict__ out_value,
                              float* __restrict__ out_w5) {
  extern __shared__ float smem[];
  float* sOA = smem;                    // 64*148
  float* sT  = sOA + Nn * OA_LD;        // 64*148
  float* sW  = sT  + Nn * OA_LD;        // 64*65
  float* sE  = sW  + Nn * W_LD;         // 64
  float* sS  = sE  + Nn;                // 64
  float* sVA = sS  + Nn;                // 64
  float* sD  = sVA + Nn;                // 64

  const int b    = blockIdx.x;
  const int tid  = threadIdx.x;
  const int lane = tid & 31;
  const int wave = tid >> 5;

  const float* __restrict__ G = ws + G_OFF;
  const float* __restrict__ u = ws + U_OFF;
  const float* __restrict__ v = ws + V_OFF;

  const float* obs_b = obs + (size_t)b * Nn * OBS;
  const float* act_b = act + (size_t)b * Nn * ACT;
  const float* pol_b = pol + (size_t)b * Nn * ACT;

  // --- load OA = [obs | act] for this batch into LDS -----------------------
  for (int idx = tid; idx < Nn * Dd; idx += 256) {
    const int j = idx / Dd, d = idx % Dd;
    sOA[j * OA_LD + d] = (d < OBS) ? obs_b[j * OBS + d]
                                   : act_b[j * ACT + (d - OBS)];
  }
  __syncthreads();

  // --- per-row scalars (wave-uniform branches; runs alongside GEMM1) -------
  if (tid < Nn) {                                   // va[j] = OA[j].v
    float acc = 0.f;
    for (int d = 0; d < Dd; ++d) acc += sOA[tid * OA_LD + d] * v[d];
    sVA[tid] = acc;
  } else if (tid < 2 * Nn) {                        // e[i] = obs[i].u
    const int i = tid - Nn;
    float acc = 0.f;
    for (int o = 0; o < OBS; ++o) acc += sOA[i * OA_LD + o] * u[o];
    sE[i] = acc;
  } else if (tid < 3 * Nn) {                        // dlt[m] = (pol-act)[m].v_act
    const int m = tid - 2 * Nn;
    float acc = 0.f;
    for (int a = 0; a < ACT; ++a)
      acc += (pol_b[m * ACT + a] - act_b[m * ACT + a]) * v[OBS + a];
    sD[m] = acc;
  }

  const int m16   = lane & 15;            // row/col within 16-tile
  const int khalf = (lane >> 4) << 1;     // 0 or 2 (K sub-chunk per half-wave)
  const int rhalf = (lane >> 4) << 3;     // 0 or 8 (C/D row offset per half-wave)

  // --- GEMM1: T = OA * G   (tiles: 4 x 9, Kdim = 144) ----------------------
  for (int tt = wave; tt < 4 * 9; tt += 8) {
    const int mt = tt / 9, nt = tt % 9;
    v8f acc = {};
    for (int k0 = 0; k0 < Dd; k0 += 4) {
      v2f a, bb;
      a.x  = sOA[(mt * 16 + m16) * OA_LD + k0 + khalf + 0];
      a.y  = sOA[(mt * 16 + m16) * OA_LD + k0 + khalf + 1];
      bb.x = G[(k0 + khalf + 0) * Dd + nt * 16 + m16];
      bb.y = G[(k0 + khalf + 1) * Dd + nt * 16 + m16];
      acc = __builtin_amdgcn_wmma_f32_16x16x4_f32(false, a, false, bb,
                                                  (short)0, acc, false, false);
    }
    #pragma unroll
    for (int p = 0; p < 8; ++p)
      sT[(mt * 16 + rhalf + p) * OA_LD + nt * 16 + m16] = acc[p];
  }
  __syncthreads();

  // --- GEMM2: scores = T * OA^T  (tiles: 4 x 4), sigmoid into sW -----------
  const float scale = 0.08838834764831845f;   // 1/sqrt(128)
  for (int tt = wave; tt < 16; tt += 8) {
    const int it = tt >> 2, jt = tt & 3;
    v8f acc = {};
    for (int k0 = 0; k0 < Dd; k0 += 4) {
      v2f a, bb;
      a.x  = sT [(it * 16 + m16) * OA_LD + k0 + khalf + 0];
      a.y  = sT [(it * 16 + m16) * OA_LD + k0 + khalf + 1];
      bb.x = sOA[(jt * 16 + m16) * OA_LD + k0 + khalf + 0];
      bb.y = sOA[(jt * 16 + m16) * OA_LD + k0 + khalf + 1];
      acc = __builtin_amdgcn_wmma_f32_16x16x4_f32(false, a, false, bb,
                                                  (short)0, acc, false, false);
    }
    #pragma unroll
    for (int p = 0; p < 8; ++p) {
      const float sc = acc[p] * scale;
      sW[(it * 16 + rhalf + p) * W_LD + jt * 16 + m16] =
          1.0f / (1.0f + __expf(-sc));
    }
  }
  __syncthreads();

  // --- s[i] = sum_j w[i,j]*va[j] -------------------------------------------
  if (tid < Nn) {
    float acc = 0.f;
    for (int j = 0; j < Nn; ++j) acc += sW[tid * W_LD + j] * sVA[j];
    sS[tid] = acc;
  }
  __syncthreads();

  // --- value[b,i,m] = e[i] + (s[i] + w[i,m]*dlt[m]) / N --------------------
  float* val_b = out_value + (size_t)b * Nn * Nn;
  const float invN = 1.0f / (float)Nn;
  for (int idx = tid; idx < Nn * Nn; idx += 256) {
    const int i = idx >> 6, m = idx & 63;
    val_b[idx] = sE[i] + (sS[i] + sW[i * W_LD + m] * sD[m]) * invN;
  }

  // --- weights5[b,i,x,m] = w[i,m] : broadcast stream, float4 stores --------
  float4* w5_b = (float4*)(out_w5 + (size_t)b * Nn * Nn * Nn);
  for (int idx4 = tid; idx4 < (Nn * Nn * Nn) / 4; idx4 += 256) {
    const int i  = idx4 >> 10;            // flat/4096
    const int m0 = (idx4 & 15) << 2;      // (flat%64) aligned to 4
    float4 val;
    val.x = sW[i * W_LD + m0 + 0];
    val.y = sW[i * W_LD + m0 + 1];
    val.z = sW[i * W_LD + m0 + 2];
    val.w = sW[i * W_LD + m0 + 3];
    w5_b[idx4] = val;
  }
}

extern "C" void kernel_launch(void* const* d_in, const int* in_sizes, int n_in,
                              void* d_out, int out_size, void* d_ws, size_t ws_size,
                              hipStream_t stream) {
  (void)in_sizes; (void)n_in; (void)out_size; (void)ws_size;
  const float* obs  = (const float*)d_in[0];
  const float* pol  = (const float*)d_in[1];
  const float* act  = (const float*)d_in[2];
  const float* Wk   = (const float*)d_in[3];
  const float* Wq   = (const float*)d_in[4];
  const float* Wv   = (const float*)d_in[5];
  const float* Wenc = (const float*)d_in[6];
  const float* Wfin = (const float*)d_in[7];

  float* ws        = (float*)d_ws;
  float* out_value = (float*)d_out;                       // [B,N,N,1] first
  float* out_w5    = out_value + (size_t)Bsz * Nn * Nn;   // then [B,N,N,N,1]

  precompute_kernel<<<Dd + 2, Dd, 0, stream>>>(Wk, Wq, Wv, Wenc, Wfin, ws);

  const size_t shmem =
      (size_t)(2 * Nn * OA_LD + Nn * W_LD + 4 * Nn) * sizeof(float);
  critic_kernel<<<Bsz, 256, shmem, stream>>>(obs, pol, act, ws,
                                             out_value, out_w5);
}